// Layer_14568529068705
// MI455X (gfx1250) — compile-verified
//
#include <hip/hip_runtime.h>
#include <math.h>

// ---------------------------------------------------------------------------
// Vanilla tanh-RNN for MI455X (gfx1250, wave32, WMMA, async-LDS).
//   T=2048, B=64, D=512, H=512, all fp32.
//
// Kernel 1 (xproj): x_proj = inp @ W_ih^T + b_ih  -> written into d_out.
//   fp32 WMMA (V_WMMA_F32_16X16X4_F32). Block = 8 waves owns a 32-row M tile
//   (staged into LDS with global_load_async_to_lds_b128) x all 512 columns;
//   each wave: 2 M-tiles x 4 N-tiles = 8 accumulators, so every B-fragment
//   load feeds two WMMAs (2x arithmetic intensity vs 16-row tiles).
//
// Kernel 2 (scan): persistent kernel, 16 blocks x 8 waves = 128 waves = the
//   128 16x16 tiles of the 64x512 hidden-state update. W_hh slice cached in
//   LDS once; h double-buffered in d_ws; device-scope atomic barrier per step.
//   outputs[t] (= x_proj[t]) is read and overwritten in place with
//   tanh(...) computed by the gfx1250 hardware v_tanh_f32.
// ---------------------------------------------------------------------------

typedef __attribute__((ext_vector_type(2))) float v2f;
typedef __attribute__((ext_vector_type(8))) float v8f;

#define T_STEPS 2048
#define B_SZ    64
#define D_SZ    512
#define H_SZ    512
#define M_TOTAL (T_STEPS * B_SZ)   // 131072 rows of the projection GEMM

#define MROWS 32  // M rows staged per block in kernel 1
#define ALD 520   // LDS row stride (floats): 520%64=8 -> <=2-way bank conflict; 16B aligned
#define WLD 520   // LDS row stride for W_hh slice
#define NB  16    // persistent blocks in the scan kernel

// D = A(16x4) * B(4x16) + C, fp32, wave32.
// Per-lane layout (ISA 7.12.2): lane L, g=L>>4, i=L&15:
//   A: a.x = A[m=i][k=2g],  a.y = A[m=i][k=2g+1]
//   B: b.x = B[k=2g][n=i],  b.y = B[k=2g+1][n=i]
//   C/D: c[j] = C[m=j+8g][n=i]
__device__ __forceinline__ v8f wmma_f32x4(v2f a, v2f b, v8f c) {
  return __builtin_amdgcn_wmma_f32_16x16x4_f32(
      /*neg_a=*/false, a, /*neg_b=*/false, b,
      /*c_mod=*/(short)0, c, /*reuse_a=*/false, /*reuse_b=*/false);
}

// gfx1250 hardware tanh. TRANS-op hazard rule (ISA 5.7.4): one independent
// op (v_nop) before the result is consumed / sources overwritten.
__device__ __forceinline__ float hw_tanhf(float x) {
  float r;
  asm volatile("v_tanh_f32 %0, %1\n\tv_nop" : "=v"(r) : "v"(x));
  return r;
}

// Async global -> LDS copy (16B), tracked by ASYNCcnt. lds must be the
// 32-bit LDS byte address (low 32 bits of the flat shared pointer: the LDS
// aperture truncates addr[31:0]).
__device__ __forceinline__ void async_copy_b128(void* lds_ptr, const void* gptr) {
  unsigned lds32 = (unsigned)(size_t)lds_ptr;
  asm volatile("global_load_async_to_lds_b128 %0, %1, off"
               :: "v"(lds32), "v"(gptr) : "memory");
}
__device__ __forceinline__ void wait_async0() {
  asm volatile("s_wait_asynccnt 0x0" ::: "memory");
}

// ---------------------------------------------------------------------------
// Kernel 1: x_proj[m, n] = sum_k inp[m, k] * W_ih[n, k] + b_ih[n]
// ---------------------------------------------------------------------------
__global__ __launch_bounds__(256)
void xproj_kernel(const float* __restrict__ inp,
                  const float* __restrict__ W_ih,
                  const float* __restrict__ b_ih,
                  float* __restrict__ xout) {
  __shared__ float Alds[MROWS * ALD];   // 32 x 512 A tile (65 KB + pad)

  const int tid  = threadIdx.x;
  const int wave = tid >> 5;
  const int lane = tid & 31;
  const int g    = lane >> 4;   // half-wave
  const int il   = lane & 15;
  const size_t mbase = (size_t)blockIdx.x * MROWS;

  // Async DMA stage of the 32x512 A tile: thread t handles row t>>3,
  // 64-float slab (t&7)*64, as 16B async copies (no VGPR data movement).
  {
    const int r  = tid >> 3;          // 0..31
    const int c0 = (tid & 7) * 64;    // 8 threads per row
    const float* src = inp + (mbase + r) * D_SZ + c0;
    float* dst = &Alds[r * ALD + c0];
#pragma unroll
    for (int i = 0; i < 64; i += 4)
      async_copy_b128(dst + i, src + i);
  }
  wait_async0();
  __syncthreads();

  const int n0 = wave * 64;  // this wave's 64-column strip
  v8f acc[2][4];
#pragma unroll
  for (int mt = 0; mt < 2; ++mt)
#pragma unroll
    for (int nt = 0; nt < 4; ++nt)
#pragma unroll
      for (int j = 0; j < 8; ++j) acc[mt][nt][j] = 0.0f;

  // B fragments straight from W_ih rows (row-major [n][k] supplies the
  // per-lane-transposed B[k][n] layout). W_ih is 1 MB -> L2/WGP$ resident.
  const float* w0 = W_ih + (size_t)(n0 +  0 + il) * D_SZ;
  const float* w1 = W_ih + (size_t)(n0 + 16 + il) * D_SZ;
  const float* w2 = W_ih + (size_t)(n0 + 32 + il) * D_SZ;
  const float* w3 = W_ih + (size_t)(n0 + 48 + il) * D_SZ;
  const float* arow0 = &Alds[il * ALD];
  const float* arow1 = &Alds[(16 + il) * ALD];

#pragma unroll 4
  for (int k = 0; k < D_SZ; k += 4) {
    const int ko = k + 2 * g;
    v2f a0 = *(const v2f*)(arow0 + ko);       // ds_read_b64
    v2f a1 = *(const v2f*)(arow1 + ko);
    v2f b0 = *(const v2f*)(w0 + ko);          // global_load_b64
    v2f b1 = *(const v2f*)(w1 + ko);
    v2f b2 = *(const v2f*)(w2 + ko);
    v2f b3 = *(const v2f*)(w3 + ko);
    acc[0][0] = wmma_f32x4(a0, b0, acc[0][0]);
    acc[1][0] = wmma_f32x4(a1, b0, acc[1][0]);
    acc[0][1] = wmma_f32x4(a0, b1, acc[0][1]);
    acc[1][1] = wmma_f32x4(a1, b1, acc[1][1]);
    acc[0][2] = wmma_f32x4(a0, b2, acc[0][2]);
    acc[1][2] = wmma_f32x4(a1, b2, acc[1][2]);
    acc[0][3] = wmma_f32x4(a0, b3, acc[0][3]);
    acc[1][3] = wmma_f32x4(a1, b3, acc[1][3]);
  }

  // Epilogue: + b_ih, scatter to d_out (row-major [m][n]).
#pragma unroll
  for (int nt = 0; nt < 4; ++nt) {
    const int n = n0 + nt * 16 + il;
    const float bias = b_ih[n];
#pragma unroll
    for (int mt = 0; mt < 2; ++mt)
#pragma unroll
      for (int j = 0; j < 8; ++j)
        xout[(mbase + mt * 16 + 8 * g + j) * H_SZ + n] = acc[mt][nt][j] + bias;
  }
}

// ---------------------------------------------------------------------------
// Device-scope barrier for the persistent scan kernel.
// ---------------------------------------------------------------------------
__device__ __forceinline__ void grid_barrier(unsigned* cnt, unsigned* gen,
                                             unsigned nblocks) {
  __syncthreads();
  if (threadIdx.x == 0) {
    __threadfence();  // make this block's h writes device-visible
    unsigned g = __hip_atomic_load(gen, __ATOMIC_RELAXED, __HIP_MEMORY_SCOPE_AGENT);
    unsigned prev = __hip_atomic_fetch_add(cnt, 1u, __ATOMIC_ACQ_REL, __HIP_MEMORY_SCOPE_AGENT);
    if (prev == nblocks - 1u) {
      __hip_atomic_store(cnt, 0u, __ATOMIC_RELAXED, __HIP_MEMORY_SCOPE_AGENT);
      __hip_atomic_store(gen, g + 1u, __ATOMIC_RELEASE, __HIP_MEMORY_SCOPE_AGENT);
    } else {
      while (__hip_atomic_load(gen, __ATOMIC_ACQUIRE, __HIP_MEMORY_SCOPE_AGENT) == g) {
        __builtin_amdgcn_s_sleep(1);
      }
    }
  }
  __syncthreads();
}

// ---------------------------------------------------------------------------
// Kernel 2: persistent recurrent scan.
//   wave -> one 16x16 tile: mtile = wave&3 (rows of B=64), ntile = blk*2 + (wave>>2).
//   h_new = tanh(xp + h @ W_hh^T + b_hh); xp read from (and result written to) d_out.
// ---------------------------------------------------------------------------
__global__ __launch_bounds__(256)
void scan_kernel(const float* __restrict__ W_hh,
                 const float* __restrict__ b_hh,
                 float* __restrict__ out,
                 float* __restrict__ ws) {
  __shared__ float Wlds[32 * WLD];   // this block's 32 rows of W_hh (65 KB)

  float* h0 = ws;                       // h double buffer in workspace
  float* h1 = ws + B_SZ * H_SZ;
  unsigned* sync = (unsigned*)(ws + 2 * B_SZ * H_SZ);

  const int tid   = threadIdx.x;
  const int wave  = tid >> 5;
  const int lane  = tid & 31;
  const int g     = lane >> 4;
  const int il    = lane & 15;
  const int mtile = wave & 3;           // 0..3  -> rows [16*mtile, +16)
  const int ntsel = wave >> 2;          // 0..1
  const int nglob = blockIdx.x * 32 + ntsel * 16 + il;   // this lane's column

  // Stage W_hh rows [blk*32, +32) into LDS once (loop-invariant) via async DMA.
  {
    const int r  = tid >> 3;            // 0..31
    const int c0 = (tid & 7) * 64;      // 8 threads per row, 64 floats each
    const float* src = W_hh + ((size_t)blockIdx.x * 32 + r) * H_SZ + c0;
    float* dst = &Wlds[r * WLD + c0];
#pragma unroll
    for (int i = 0; i < 64; i += 4)
      async_copy_b128(dst + i, src + i);
  }
  wait_async0();
  __syncthreads();

  const float bias = b_hh[nglob];
  const float* bptr = &Wlds[(ntsel * 16 + il) * WLD];

  for (int t = 0; t < T_STEPS; ++t) {
    const float* hc = (t & 1) ? h1 : h0;
    float*       hn = (t & 1) ? h0 : h1;
    const float* arow = hc + (size_t)(mtile * 16 + il) * H_SZ;  // L2-resident (128 KB)

    v8f c;
#pragma unroll
    for (int j = 0; j < 8; ++j) c[j] = 0.0f;

#pragma unroll 8
    for (int k = 0; k < H_SZ; k += 4) {
      const int ko = k + 2 * g;
      v2f a = *(const v2f*)(arow + ko);   // global_load_b64 (h)
      v2f b = *(const v2f*)(bptr + ko);   // ds_read_b64 (W_hh)
      c = wmma_f32x4(a, b, c);
    }

    const size_t obase = (size_t)t * (B_SZ * H_SZ);
#pragma unroll
    for (int j = 0; j < 8; ++j) {
      const int row = mtile * 16 + 8 * g + j;
      const size_t idx = obase + (size_t)row * H_SZ + nglob;
      const float v = hw_tanhf(c[j] + out[idx] + bias);  // out[idx] holds x_proj
      out[idx] = v;                                      // overwrite with h_t
      hn[row * H_SZ + nglob] = v;                        // next step's input
    }
    grid_barrier(&sync[0], &sync[1], NB);
  }

  // Final state = h after T steps; T even -> lives in h0. Cooperative copy.
  float* fs = out + (size_t)T_STEPS * B_SZ * H_SZ;
  for (int i = blockIdx.x * 256 + tid; i < B_SZ * H_SZ; i += NB * 256)
    fs[i] = h0[i];
}

// ---------------------------------------------------------------------------
// Init: h0 <- state, reset barrier words (d_ws is poisoned / left dirty).
// ---------------------------------------------------------------------------
__global__ void init_kernel(const float* __restrict__ state, float* __restrict__ ws) {
  const int i = blockIdx.x * blockDim.x + threadIdx.x;
  if (i < B_SZ * H_SZ) ws[i] = state[i];
  if (i < 2) ((unsigned*)(ws + 2 * B_SZ * H_SZ))[i] = 0u;
}

extern "C" void kernel_launch(void* const* d_in, const int* in_sizes, int n_in,
                              void* d_out, int out_size, void* d_ws, size_t ws_size,
                              hipStream_t stream) {
  const float* inp   = (const float*)d_in[0];
  const float* state = (const float*)d_in[1];
  const float* W_ih  = (const float*)d_in[2];
  const float* W_hh  = (const float*)d_in[3];
  const float* b_ih  = (const float*)d_in[4];
  const float* b_hh  = (const float*)d_in[5];
  float* out = (float*)d_out;
  float* ws  = (float*)d_ws;

  init_kernel<<<(B_SZ * H_SZ + 255) / 256, 256, 0, stream>>>(state, ws);
  xproj_kernel<<<M_TOTAL / MROWS, 256, 0, stream>>>(inp, W_ih, b_ih, out);
  scan_kernel<<<NB, 256, 0, stream>>>(W_hh, b_hh, out, ws);
}